// PixelDSNTDistanceDoubleEval_55035710931299
// MI455X (gfx1250) — compile-verified
//
#include <hip/hip_runtime.h>

typedef float v2f __attribute__((ext_vector_type(2)));
typedef float v4f __attribute__((ext_vector_type(4)));
typedef float v8f __attribute__((ext_vector_type(8)));

#define HW      (512 * 512)
#define WDIM    512
#define BC      128
#define THREADS 256
#define ELEMS   8
#define CHUNK   (THREADS * 4 * ELEMS)   /* 8192 elements per block */
#define NCHUNK  (HW / CHUNK)            /* 32 blocks per (b,c) pair */

// Monotonic unsigned mapping of float: preserves ordering so integer max == float max.
__device__ __forceinline__ unsigned monokey(float f) {
    unsigned u = __float_as_uint(f);
    return u ^ ((u & 0x80000000u) ? 0xFFFFFFFFu : 0x80000000u);
}

__global__ void init_ws(float* fs, unsigned long long* am) {
    int i = threadIdx.x;
    if (i < 3 * BC) fs[i] = 0.0f;
    if (i < BC)     am[i] = 0ull;
}

__global__ __launch_bounds__(THREADS)
void dsnt_stats(const float* __restrict__ inp, const float* __restrict__ tgt,
                float* __restrict__ g_se, float* __restrict__ g_sx,
                float* __restrict__ g_sy, unsigned long long* __restrict__ g_am) {
    const int pair  = blockIdx.y;
    const int chunk = blockIdx.x;
    const size_t pbase = (size_t)pair * HW;

    float se = 0.0f, sx = 0.0f, sy = 0.0f;
    float bv = -__builtin_huge_valf();
    unsigned bi = 0u;

#pragma unroll
    for (int e = 0; e < ELEMS; e++) {
        int local = chunk * CHUNK + (e * THREADS + threadIdx.x) * 4;
        // One-pass stream, zero reuse -> non-temporal b128 loads (TH_NT).
        v4f v = __builtin_nontemporal_load((const v4f*)(inp + pbase + local));
        v4f g = __builtin_nontemporal_load((const v4f*)(tgt + pbase + local));

        // coordinates: x = (local % 512) + 1 (+0..3), y = local/512 + 1 (row-aligned float4)
        float xb = (float)((local & (WDIM - 1)) + 1);
        float yv = (float)((local >> 9) + 1);

        float e0 = __expf(v.x), e1 = __expf(v.y), e2 = __expf(v.z), e3 = __expf(v.w);
        float es = e0 + e1 + e2 + e3;
        se += es;
        // sum e_k*(xb+k) = es*xb + (e1 + 2*e2 + 3*e3)
        float xoff = fmaf(3.0f, e3, fmaf(2.0f, e2, e1));
        sx += fmaf(es, xb, xoff);
        sy += es * yv;

        // argmax tracking: strict '>' keeps the first (lowest) index per thread
        if (g.x > bv) { bv = g.x; bi = (unsigned)(local + 0); }
        if (g.y > bv) { bv = g.y; bi = (unsigned)(local + 1); }
        if (g.z > bv) { bv = g.z; bi = (unsigned)(local + 2); }
        if (g.w > bv) { bv = g.w; bi = (unsigned)(local + 3); }
    }

    // pack (value, ~index) once per thread: max key == max value, lowest index on ties
    unsigned long long best = ((unsigned long long)monokey(bv) << 32) | (unsigned)~bi;

    // wave32 butterfly reduction
    for (int off = 16; off > 0; off >>= 1) {
        se += __shfl_xor(se, off, 32);
        sx += __shfl_xor(sx, off, 32);
        sy += __shfl_xor(sy, off, 32);
        unsigned long long o = __shfl_xor(best, off, 32);
        if (o > best) best = o;
    }

    __shared__ float ws[8][3];
    __shared__ unsigned long long wk[8];
    int wave = threadIdx.x >> 5;
    int lane = threadIdx.x & 31;
    if (lane == 0) { ws[wave][0] = se; ws[wave][1] = sx; ws[wave][2] = sy; wk[wave] = best; }
    __syncthreads();

    if (threadIdx.x < 32) {
        // Combine the 8 wave partials for all 3 quantities with one WMMA:
        // D = A x ones, D[m][j] = sum_k A[m][k].  A 16x4 fp32 layout:
        // lane L(0..15) holds A[L][0],A[L][1]; lane L+16 holds A[L][2],A[L][3].
        v2f a; a.x = 0.0f; a.y = 0.0f;
        int m = lane & 15;
        if (m < 3) {
            if (lane < 16) { a.x = ws[0][m] + ws[1][m]; a.y = ws[2][m] + ws[3][m]; }
            else           { a.x = ws[4][m] + ws[5][m]; a.y = ws[6][m] + ws[7][m]; }
        }
        v2f bo; bo.x = 1.0f; bo.y = 1.0f;   // all-ones B: layout-independent
        v8f c = {};
        c = __builtin_amdgcn_wmma_f32_16x16x4_f32(false, a, false, bo,
                                                  (short)0, c, false, false);
        // D layout: VGPR r, lanes 0-15 = D[r][lane]; lane0 sees rows 0..7 at col 0.
        if (lane == 0) {
            atomicAdd(&g_se[pair], c[0]);
            atomicAdd(&g_sx[pair], c[1]);
            atomicAdd(&g_sy[pair], c[2]);
            unsigned long long bk = wk[0];
            for (int i = 1; i < 8; i++) if (wk[i] > bk) bk = wk[i];
            atomicMax(&g_am[pair], bk);
        }
    }
}

__global__ __launch_bounds__(64)
void dsnt_final(const float* __restrict__ g_se, const float* __restrict__ g_sx,
                const float* __restrict__ g_sy, const unsigned long long* __restrict__ g_am,
                float* __restrict__ out) {
    __shared__ float sed0[64], sed1[64], sdd[64];
    const int b = threadIdx.x;
    {
        int p0 = b * 2, p1 = b * 2 + 1;
        float px0 = g_sx[p0] / g_se[p0], py0 = g_sy[p0] / g_se[p0];
        float px1 = g_sx[p1] / g_se[p1], py1 = g_sy[p1] / g_se[p1];
        unsigned i0 = ~(unsigned)(g_am[p0] & 0xFFFFFFFFull);
        unsigned i1 = ~(unsigned)(g_am[p1] & 0xFFFFFFFFull);
        float tx0 = (float)((i0 & 511u) + 1u), ty0 = (float)((i0 >> 9) + 1u);
        float tx1 = (float)((i1 & 511u) + 1u), ty1 = (float)((i1 >> 9) + 1u);
        float dx0 = tx0 - px0, dy0 = ty0 - py0;
        float dx1 = tx1 - px1, dy1 = ty1 - py1;
        sed0[b] = sqrtf(dx0 * dx0 + dy0 * dy0);
        sed1[b] = sqrtf(dx1 * dx1 + dy1 * dy1);
        float vpx = px0 - px1, vpy = py0 - py1, vtx = tx0 - tx1, vty = ty0 - ty1;
        float pd = sqrtf(vpx * vpx + vpy * vpy);
        float td = sqrtf(vtx * vtx + vty * vty);
        sdd[b] = fabsf(pd - td);
    }
    __syncthreads();

    if (threadIdx.x < 32) {
        int lane = threadIdx.x;
        v2f bo; bo.x = 1.0f; bo.y = 1.0f;
        float tot[3];
        const float* arrs[3] = { sed0, sed1, sdd };
#pragma unroll
        for (int q = 0; q < 3; q++) {
            const float* v = arrs[q];
            v2f a;
            int m = lane & 15;
            if (lane < 16) { a.x = v[4 * m + 0]; a.y = v[4 * m + 1]; }
            else           { a.x = v[4 * m + 2]; a.y = v[4 * m + 3]; }
            v8f c = {};
            c = __builtin_amdgcn_wmma_f32_16x16x4_f32(false, a, false, bo,
                                                      (short)0, c, false, false);
            // lanes 0-15 hold rows 0-7 at their column; lanes 16-31 hold rows 8-15
            float p = c[0] + c[1] + c[2] + c[3] + c[4] + c[5] + c[6] + c[7];
            p += __shfl_xor(p, 16, 32);
            tot[q] = p;
        }
        if (lane == 0) {
            float si = tot[0], ss = tot[1], sd = tot[2];
            out[0] = si / 64.0f;
            out[1] = ss / 64.0f;
            out[2] = (si + ss) / 64.0f;
            out[3] = sd / 64.0f;
        }
    }
}

extern "C" void kernel_launch(void* const* d_in, const int* in_sizes, int n_in,
                              void* d_out, int out_size, void* d_ws, size_t ws_size,
                              hipStream_t stream) {
    (void)in_sizes; (void)n_in; (void)out_size; (void)ws_size;
    const float* inp = (const float*)d_in[0];
    const float* tgt = (const float*)d_in[1];

    float* fs = (float*)d_ws;
    float* g_se = fs;
    float* g_sx = fs + BC;
    float* g_sy = fs + 2 * BC;
    unsigned long long* g_am = (unsigned long long*)(fs + 3 * BC);  // offset 1536B, 8B aligned

    init_ws<<<1, 512, 0, stream>>>(fs, g_am);

    dim3 grid(NCHUNK, BC);
    dsnt_stats<<<grid, THREADS, 0, stream>>>(inp, tgt, g_se, g_sx, g_sy, g_am);

    dsnt_final<<<1, 64, 0, stream>>>(g_se, g_sx, g_sy, g_am, (float*)d_out);
}